// RWKV_63960652972091
// MI455X (gfx1250) — compile-verified
//
#include <hip/hip_runtime.h>
#include <hip/hip_bf16.h>
#include <math.h>

#define D 1024
#define L 24
#define V 50277
#define FF 4096
#define KSLICE 128

typedef __attribute__((ext_vector_type(2))) float v2f;
typedef __attribute__((ext_vector_type(8))) float v8f;

// ---------------- block-wide sum over 1024 threads (32 waves, wave32) -------
__device__ __forceinline__ float block_sum_1024(float v, float* sbuf) {
  const int tid = threadIdx.x;
#pragma unroll
  for (int off = 16; off > 0; off >>= 1) v += __shfl_down(v, off, 32);
  if ((tid & 31) == 0) sbuf[tid >> 5] = v;
  __syncthreads();
  if (tid < 32) {
    float w = sbuf[tid];
#pragma unroll
    for (int off = 16; off > 0; off >>= 1) w += __shfl_down(w, off, 32);
    if (tid == 0) sbuf[0] = w;
  }
  __syncthreads();
  float r = sbuf[0];
  __syncthreads();
  return r;
}

// ---------------- layernorm of a D-vector (one block of 1024) ---------------
__global__ void k_ln(const float* __restrict__ in, const float* __restrict__ w,
                     const float* __restrict__ b, float* __restrict__ out) {
  __shared__ float sbuf[32];
  const int i = threadIdx.x;
  float xi = in[i];
  float mu = block_sum_1024(xi, sbuf) * (1.0f / D);
  float d  = xi - mu;
  float var = block_sum_1024(d * d, sbuf) * (1.0f / D);
  out[i] = d * rsqrtf(var + 1e-5f) * w[i] + b[i];
}

// ---------------- attention premix: LN1 + token-mix, zero gemv outputs ------
__global__ void k_att_premix(const float* __restrict__ x,
                             const float* __restrict__ w, const float* __restrict__ b,
                             const float* __restrict__ prev,
                             const float* __restrict__ tmk, const float* __restrict__ tmv,
                             const float* __restrict__ tmr,
                             float* __restrict__ xk, float* __restrict__ xv,
                             float* __restrict__ xr,
                             float* __restrict__ kk, float* __restrict__ vv,
                             float* __restrict__ rr,
                             float* __restrict__ xln_out) {
  __shared__ float sbuf[32];
  const int i = threadIdx.x;
  float xi = x[i];
  float mu = block_sum_1024(xi, sbuf) * (1.0f / D);
  float d  = xi - mu;
  float var = block_sum_1024(d * d, sbuf) * (1.0f / D);
  float xl = d * rsqrtf(var + 1e-5f) * w[i] + b[i];
  xln_out[i] = xl;                       // new state row 5l+1
  float pv = prev[i];
  float km = tmk[i], vm = tmv[i], rm = tmr[i];
  xk[i] = xl * km + pv * (1.0f - km);
  xv[i] = xl * vm + pv * (1.0f - vm);
  xr[i] = xl * rm + pv * (1.0f - rm);
  kk[i] = 0.0f; vv[i] = 0.0f; rr[i] = 0.0f;
}

// ---------------- WKV recurrence (elementwise over D) -----------------------
__global__ void k_wkv(const float* __restrict__ kk, const float* __restrict__ vv,
                      const float* __restrict__ rr,
                      const float* __restrict__ tf, const float* __restrict__ td,
                      const float* __restrict__ aaS, const float* __restrict__ bbS,
                      const float* __restrict__ ppS,
                      float* __restrict__ wkv,
                      float* __restrict__ naO, float* __restrict__ nbO,
                      float* __restrict__ p2O, float* __restrict__ att_out) {
  const int i = threadIdx.x;
  float k = kk[i], v = vv[i];
  float r = 1.0f / (1.0f + expf(-rr[i]));
  float aa = aaS[i], bb = bbS[i], pp = ppS[i];
  float ww = tf[i] + k;
  float p1 = fmaxf(pp, ww);
  float e1 = expf(pp - p1), e2 = expf(ww - p1);
  float a = e1 * aa + e2 * v;
  float b = e1 * bb + e2;
  float ww2 = pp + td[i];
  float p2 = fmaxf(ww2, k);
  float f1 = expf(ww2 - p2), f2 = expf(k - p2);
  naO[i] = f1 * aa + f2 * v;             // state row 5l+2
  nbO[i] = f1 * bb + f2;                 // state row 5l+3
  p2O[i] = p2;                           // state row 5l+4
  wkv[i] = r * (a / b);
  att_out[i] = 0.0f;                     // zero next gemv accumulator
}

// ---------------- ffn premix: x += att_out, LN2, mixes, zero accumulators ---
__global__ void k_ffn_premix(float* __restrict__ x, const float* __restrict__ att_out,
                             const float* __restrict__ w, const float* __restrict__ b,
                             const float* __restrict__ prev0,
                             const float* __restrict__ ftmk, const float* __restrict__ ftmr,
                             float* __restrict__ fxk, float* __restrict__ fxr,
                             float* __restrict__ frr, float* __restrict__ kch,
                             float* __restrict__ fvk, float* __restrict__ xln2_out) {
  __shared__ float sbuf[32];
  const int i = threadIdx.x;
  float xi = x[i] + att_out[i];
  x[i] = xi;
  float mu = block_sum_1024(xi, sbuf) * (1.0f / D);
  float d  = xi - mu;
  float var = block_sum_1024(d * d, sbuf) * (1.0f / D);
  float xl = d * rsqrtf(var + 1e-5f) * w[i] + b[i];
  xln2_out[i] = xl;                      // new state row 5l+0
  float pv = prev0[i];
  float km = ftmk[i], rm = ftmr[i];
  fxk[i] = xl * km + pv * (1.0f - km);
  fxr[i] = xl * rm + pv * (1.0f - rm);
  frr[i] = 0.0f; fvk[i] = 0.0f;
  kch[i] = 0.0f; kch[i + D] = 0.0f; kch[i + 2 * D] = 0.0f; kch[i + 3 * D] = 0.0f;
}

// ---------------- ffn epilogue: x = (x + sigmoid(frr)*fvk) * sc -------------
__global__ void k_ffn_out(float* __restrict__ x, const float* __restrict__ frr,
                          const float* __restrict__ fvk, float sc) {
  const int i = threadIdx.x;
  float r = 1.0f / (1.0f + expf(-frr[i]));
  x[i] = (x[i] + r * fvk[i]) * sc;
}

__global__ void k_zero(float* __restrict__ p, int n) {
  int i = blockIdx.x * blockDim.x + threadIdx.x;
  if (i < n) p[i] = 0.0f;
}

// ---------------- WMMA GEMV: y += W(MxK,row-major) @ f(x) -------------------
// A = 16x4 fp32 tile of W (ISA layout: lanes 0-15 -> k0,k0+1; lanes 16-31 ->
// k0+2,k0+3), B = x broadcast across N. Column N=0 of D (lanes 0/16) holds the
// 16 partial dot products; split-K partials combined with global f32 atomics.
// XFORM==1 applies relu(x)^2 to the x operand (RWKV channel-mix).
// x is fetched as a wave-uniform float4 per step -> scalar (SMEM) path,
// leaving the VMEM pipe to the weight stream (one b64 per lane per wmma).
template <int XFORM>
__global__ void k_gemv_wmma(const float* __restrict__ W, const float* __restrict__ xv,
                            float* __restrict__ y, int M, int K) {
  const int lane = threadIdx.x & 31;
  const int wave = threadIdx.x >> 5;
  const int m0 = blockIdx.x * 128 + wave * 16;
  if (m0 >= M) return;                   // wave-uniform: EXEC stays all-ones
  const int half = lane >> 4;            // 0: k pair {0,1}; 1: k pair {2,3}
  const int r16  = lane & 15;
  int row = m0 + r16;
  if (row >= M) row = m0;                // clamp loads; lands in unstored D rows
  const float* __restrict__ wrow = W + (size_t)row * K + 2 * half;
  const int k0 = blockIdx.y * KSLICE;
  const bool hi = (half != 0);

  v8f c = {0.f, 0.f, 0.f, 0.f, 0.f, 0.f, 0.f, 0.f};

  for (int k = k0; k < k0 + KSLICE; k += 32) {
    __builtin_prefetch(wrow + k + 512, 0, 1);   // global_prefetch_b8, ~2KB ahead
#pragma unroll
    for (int ks = 0; ks < 32; ks += 4) {
      const int kc = k + ks;
      v2f a = *(const v2f*)(wrow + kc);         // W[row][kc+2*half .. +1], b64 load
      const float4 xc = *(const float4*)(xv + kc);  // uniform -> scalar load
      float b0 = hi ? xc.y : xc.x;
      float b1 = hi ? xc.w : xc.z;
      if (XFORM == 1) {
        b0 = fmaxf(b0, 0.0f); b0 *= b0;
        b1 = fmaxf(b1, 0.0f); b1 *= b1;
      }
      v2f bv; bv.x = b0; bv.y = b1;
      c = __builtin_amdgcn_wmma_f32_16x16x4_f32(
              /*neg_a=*/false, a, /*neg_b=*/false, bv,
              /*c_mod=*/(short)0, c, /*reuse_a=*/false, /*reuse_b=*/false);
    }
  }

  if (r16 == 0) {                        // lanes 0 and 16: column N=0 of D
#pragma unroll
    for (int r = 0; r < 8; ++r) {
      int orow = m0 + half * 8 + r;
      if (orow < M) atomicAdd(&y[orow], c[r]);
    }
  }
}

// ---------------------------------------------------------------------------
extern "C" void kernel_launch(void* const* d_in, const int* in_sizes, int n_in,
                              void* d_out, int out_size, void* d_ws, size_t ws_size,
                              hipStream_t stream) {
  const float* token_embd = (const float*)d_in[0];
  const float* state      = (const float*)d_in[1];
  const float* ln0_w      = (const float*)d_in[2];
  const float* ln0_b      = (const float*)d_in[3];
  const float* ln1_w      = (const float*)d_in[4];
  const float* ln1_b      = (const float*)d_in[5];
  const float* ln2_w      = (const float*)d_in[6];
  const float* ln2_b      = (const float*)d_in[7];
  const float* att_tmk    = (const float*)d_in[8];
  const float* att_tmv    = (const float*)d_in[9];
  const float* att_tmr    = (const float*)d_in[10];
  const float* time_first = (const float*)d_in[11];
  const float* time_decay = (const float*)d_in[12];
  const float* att_kw     = (const float*)d_in[13];
  const float* att_vw     = (const float*)d_in[14];
  const float* att_rw     = (const float*)d_in[15];
  const float* att_ow     = (const float*)d_in[16];
  const float* ffn_tmk    = (const float*)d_in[17];
  const float* ffn_tmr    = (const float*)d_in[18];
  const float* ffn_kw     = (const float*)d_in[19];
  const float* ffn_vw     = (const float*)d_in[20];
  const float* ffn_rw     = (const float*)d_in[21];
  const float* lnout_w    = (const float*)d_in[22];
  const float* lnout_b    = (const float*)d_in[23];
  const float* head_w     = (const float*)d_in[24];

  float* logits = (float*)d_out;         // V
  float* stOut  = logits + V;            // 5*L*D new state

  float* ws   = (float*)d_ws;
  float* x    = ws;
  float* xk   = ws + 1 * D;
  float* xv   = ws + 2 * D;
  float* xr   = ws + 3 * D;
  float* kk   = ws + 4 * D;
  float* vv   = ws + 5 * D;
  float* rr   = ws + 6 * D;
  float* wkv  = ws + 7 * D;
  float* aout = ws + 8 * D;
  float* fxk  = ws + 9 * D;
  float* fxr  = ws + 10 * D;
  float* frr  = ws + 11 * D;
  float* fvk  = ws + 12 * D;
  float* xfin = ws + 13 * D;
  float* kch  = ws + 14 * D;             // FF floats

  const dim3 b1024(1024);
  const dim3 b256(256);
  const dim3 gDD(D / 128, D / KSLICE);       // 1024x1024 gemv
  const dim3 gFD(FF / 128, D / KSLICE);      // 4096x1024 gemv
  const dim3 gDF(D / 128, FF / KSLICE);      // 1024x4096 gemv
  const dim3 gHD((V + 127) / 128, D / KSLICE);

  k_ln<<<1, b1024, 0, stream>>>(token_embd, ln0_w, ln0_b, x);

  for (int l = 0; l < L; ++l) {
    const float* l1w  = ln1_w + l * D;
    const float* l1b  = ln1_b + l * D;
    const float* l2w  = ln2_w + l * D;
    const float* l2b  = ln2_b + l * D;
    const float* tmk  = att_tmk + l * D;
    const float* tmv  = att_tmv + l * D;
    const float* tmr  = att_tmr + l * D;
    const float* tf   = time_first + l * D;
    const float* td   = time_decay + l * D;
    const float* ftmk = ffn_tmk + l * D;
    const float* ftmr = ffn_tmr + l * D;
    const float* kw   = att_kw + (size_t)l * D * D;
    const float* vw   = att_vw + (size_t)l * D * D;
    const float* rwm  = att_rw + (size_t)l * D * D;
    const float* ow   = att_ow + (size_t)l * D * D;
    const float* fkw  = ffn_kw + (size_t)l * FF * D;
    const float* fvw  = ffn_vw + (size_t)l * D * FF;
    const float* frw  = ffn_rw + (size_t)l * D * D;

    const float* stFfn = state + (size_t)(5 * l + 0) * D;
    const float* stAtt = state + (size_t)(5 * l + 1) * D;
    const float* stAA  = state + (size_t)(5 * l + 2) * D;
    const float* stBB  = state + (size_t)(5 * l + 3) * D;
    const float* stPP  = state + (size_t)(5 * l + 4) * D;
    float* soXln2 = stOut + (size_t)(5 * l + 0) * D;
    float* soXln  = stOut + (size_t)(5 * l + 1) * D;
    float* soNA   = stOut + (size_t)(5 * l + 2) * D;
    float* soNB   = stOut + (size_t)(5 * l + 3) * D;
    float* soP2   = stOut + (size_t)(5 * l + 4) * D;

    // ---- time mixing ----
    k_att_premix<<<1, b1024, 0, stream>>>(x, l1w, l1b, stAtt, tmk, tmv, tmr,
                                          xk, xv, xr, kk, vv, rr, soXln);
    k_gemv_wmma<0><<<gDD, b256, 0, stream>>>(kw,  xk, kk, D, D);
    k_gemv_wmma<0><<<gDD, b256, 0, stream>>>(vw,  xv, vv, D, D);
    k_gemv_wmma<0><<<gDD, b256, 0, stream>>>(rwm, xr, rr, D, D);
    k_wkv<<<1, b1024, 0, stream>>>(kk, vv, rr, tf, td, stAA, stBB, stPP,
                                   wkv, soNA, soNB, soP2, aout);
    k_gemv_wmma<0><<<gDD, b256, 0, stream>>>(ow, wkv, aout, D, D);

    // ---- channel mixing ----
    k_ffn_premix<<<1, b1024, 0, stream>>>(x, aout, l2w, l2b, stFfn, ftmk, ftmr,
                                          fxk, fxr, frr, kch, fvk, soXln2);
    k_gemv_wmma<0><<<gDD, b256, 0, stream>>>(frw, fxr, frr, D, D);
    k_gemv_wmma<0><<<gFD, b256, 0, stream>>>(fkw, fxk, kch, FF, D);
    k_gemv_wmma<1><<<gDF, b256, 0, stream>>>(fvw, kch, fvk, D, FF); // relu^2 on x
    const float sc = ((l + 1) % 6 == 0) ? 0.5f : 1.0f;
    k_ffn_out<<<1, b1024, 0, stream>>>(x, frr, fvk, sc);
  }

  // ---- head ----
  k_ln<<<1, b1024, 0, stream>>>(x, lnout_w, lnout_b, xfin);
  k_zero<<<(V + 255) / 256, b256, 0, stream>>>(logits, V);
  k_gemv_wmma<0><<<gHD, b256, 0, stream>>>(head_w, xfin, logits, V, D);
}